// ProposalLayerModule_53815940219163
// MI455X (gfx1250) — compile-verified
//
#include <hip/hip_runtime.h>
#include <hip/hip_bf16.h>

// ---------------- problem constants (fixed by setup_inputs) ----------------
#define A_NUM    9
#define H_FEAT   128
#define W_FEAT   192
#define HW_FEAT  (H_FEAT * W_FEAT)          // 24576
#define N_PROP   (A_NUM * HW_FEAT)          // 221184
#define NPAD     262144                     // next pow2 >= N_PROP
#define PRE_TOPN 6000
#define POST_TOPN 300
#define TILES    (PRE_TOPN / 16)            // 375
#define NMS_TH   0.7f

typedef float v2f __attribute__((ext_vector_type(2)));
typedef float v8f __attribute__((ext_vector_type(8)));

// generate_anchors(base=16, ratios=[0.5,1,2], scales=[8,16,32]) precomputed
__constant__ float c_anchors[A_NUM][4] = {
    {-84.f,  -40.f,  99.f,  55.f},
    {-176.f, -88.f,  191.f, 103.f},
    {-360.f, -184.f, 375.f, 199.f},
    {-56.f,  -56.f,  71.f,  71.f},
    {-120.f, -120.f, 135.f, 135.f},
    {-248.f, -248.f, 263.f, 263.f},
    {-36.f,  -80.f,  51.f,  95.f},
    {-80.f,  -168.f, 95.f,  183.f},
    {-168.f, -344.f, 183.f, 359.f}};

// ---------------------------------------------------------------------------
// Kernel 1: decode boxes, compute sort keys.
// key = (~orderable(masked_score) << 32) | index  -> ascending u64 sort gives
// descending score with ascending-index tiebreak (matches jax.lax.top_k).
// ---------------------------------------------------------------------------
__global__ void prop_decode_kernel(const float* __restrict__ scores,
                                   const float* __restrict__ deltas,
                                   const float* __restrict__ iminfo,
                                   unsigned long long* __restrict__ keys,
                                   float4* __restrict__ boxesAll) {
    unsigned r = blockIdx.x * blockDim.x + threadIdx.x;
    if (r >= NPAD) return;
    if (r >= N_PROP) { keys[r] = ~0ull; return; }

    int a = (int)(r % A_NUM);
    unsigned p = r / A_NUM;
    // anchor shift ordering (meshgrid indexing='ij'): k = xi*H + yi
    int xi = (int)(p / H_FEAT), yi = (int)(p % H_FEAT);
    // delta/score ordering: p = h*W + w
    int h = (int)(p / W_FEAT), w = (int)(p % W_FEAT);

    float sx = (float)xi * 16.f, sy = (float)yi * 16.f;
    float ax1 = c_anchors[a][0] + sx, ay1 = c_anchors[a][1] + sy;
    float ax2 = c_anchors[a][2] + sx, ay2 = c_anchors[a][3] + sy;
    float wa = ax2 - ax1 + 1.f, ha = ay2 - ay1 + 1.f;
    float cx = ax1 + 0.5f * wa, cy = ay1 + 0.5f * ha;

    int base = h * W_FEAT + w;
    const float* dp = deltas + (size_t)(4 * a) * HW_FEAT + base;
    __builtin_prefetch(dp + 64, 0, 1);   // global_prefetch_b8
    float dx  = dp[0 * HW_FEAT];
    float dy  = dp[1 * HW_FEAT];
    float dw_ = dp[2 * HW_FEAT];
    float dh_ = dp[3 * HW_FEAT];
    float sc  = scores[(size_t)(A_NUM + a) * HW_FEAT + base];

    float imH = iminfo[0], imW = iminfo[1], iscale = iminfo[2];

    float pcx = dx * wa + cx, pcy = dy * ha + cy;
    float pw = expf(dw_) * wa, ph = expf(dh_) * ha;
    float x1 = pcx - 0.5f * pw, x2 = pcx + 0.5f * pw;
    float y1 = pcy - 0.5f * ph, y2 = pcy + 0.5f * ph;
    x1 = fmaxf(0.f, fminf(x1, imW - 1.f));
    x2 = fmaxf(0.f, fminf(x2, imW - 1.f));
    y1 = fmaxf(0.f, fminf(y1, imH - 1.f));
    y2 = fmaxf(0.f, fminf(y2, imH - 1.f));

    float ms = 16.f * iscale;
    bool valid = ((x2 - x1 + 1.f) >= ms) && ((y2 - y1 + 1.f) >= ms);
    float msc = valid ? sc : -__builtin_inff();

    unsigned u = __float_as_uint(msc);
    unsigned uasc = (u & 0x80000000u) ? ~u : (u | 0x80000000u);
    unsigned udesc = ~uasc;
    keys[r] = ((unsigned long long)udesc << 32) | (unsigned long long)r;
    boxesAll[r] = make_float4(x1, y1, x2, y2);
}

// ---------------------------------------------------------------------------
// Kernel 2a: one global bitonic compare-exchange step (for j >= 1024).
// ---------------------------------------------------------------------------
__global__ void bitonic_step_kernel(unsigned long long* __restrict__ keys,
                                    unsigned j, unsigned k) {
    unsigned i = blockIdx.x * blockDim.x + threadIdx.x;
    unsigned ixj = i ^ j;
    if (ixj > i) {
        unsigned long long a = keys[i], b = keys[ixj];
        bool up = ((i & k) == 0u);
        if ((a > b) == up) { keys[i] = b; keys[ixj] = a; }
    }
}

// ---------------------------------------------------------------------------
// Kernel 2b: fused bitonic stages j = j0 .. 1 inside one 1024-key LDS chunk.
// 512 threads per block; each thread performs one compare-exchange per stage.
// ---------------------------------------------------------------------------
__global__ void __launch_bounds__(512)
bitonic_fused_kernel(unsigned long long* __restrict__ keys,
                     unsigned j0, unsigned k) {
    __shared__ unsigned long long sk[1024];
    unsigned base = blockIdx.x * 1024u;
    unsigned t = threadIdx.x;

    sk[t] = keys[base + t];
    sk[t + 512] = keys[base + t + 512];
    __syncthreads();

    for (unsigned j = j0; j > 0; j >>= 1) {
        // insert a zero bit at position log2(j): i has bit j clear
        unsigned i = ((t & ~(j - 1u)) << 1) | (t & (j - 1u));
        unsigned ixj = i | j;
        bool up = (((base + i) & k) == 0u);
        unsigned long long a = sk[i], b = sk[ixj];
        if ((a > b) == up) { sk[i] = b; sk[ixj] = a; }
        __syncthreads();
    }

    keys[base + t] = sk[t];
    keys[base + t + 512] = sk[t + 512];
}

// ---------------------------------------------------------------------------
// Kernel 3: gather top-6000 boxes + validity flags.
// ---------------------------------------------------------------------------
__global__ void gather_kernel(const unsigned long long* __restrict__ keys,
                              const float4* __restrict__ boxesAll,
                              float4* __restrict__ topBoxes,
                              int* __restrict__ validFlags) {
    int t = blockIdx.x * blockDim.x + threadIdx.x;
    if (t >= PRE_TOPN) return;
    unsigned long long kk = keys[t];
    unsigned idx = (unsigned)kk;
    unsigned hi = (unsigned)(kk >> 32);
    int valid = (hi < 0xFF800000u) && (idx < N_PROP);  // -inf score -> 0xFF800000
    float4 b = (idx < N_PROP) ? boxesAll[idx] : make_float4(0.f, 0.f, 0.f, 0.f);
    topBoxes[t] = b;
    validFlags[t] = valid;
}

// ---------------------------------------------------------------------------
// Kernel 4: suppression bitmask, one wave32 per 16x16 tile (colTile >= rowTile).
//  - 32 boxes staged via gfx1250 async global->LDS copy (1 per lane).
//  - area_i + area_j tile computed on the matrix unit:
//      C = A(16x4) * B(4x16),  A[:,0]=area_row, A[:,1]=1, B[0,:]=1, B[1,:]=area_col
//    C layout: lane = column N, VGPR v = row M (+8 for lanes 16..31).
//  - IoU predicate evaluated as inter > TH * max(sum - inter, 1e-9): denom > 0,
//    so this is equivalent to inter/denom > TH but avoids the IEEE divide chain.
//  - per-lane predicate + __ballot builds two 16-bit suppression words per v.
// ---------------------------------------------------------------------------
__global__ void __launch_bounds__(32)
supp_tile_kernel(const float4* __restrict__ topBoxes,
                 unsigned* __restrict__ supp) {
    int rt = blockIdx.y, ct = blockIdx.x;
    if (ct < rt) return;
    int lane = threadIdx.x;

    __shared__ float4 sb[32];   // [0..15] row boxes, [16..31] col boxes

    int boxIdx = (lane < 16) ? (rt * 16 + lane) : (ct * 16 + (lane - 16));
    unsigned long long gaddr =
        (unsigned long long)(const void*)(topBoxes + boxIdx);
    // generic LDS pointer: low 32 bits are the workgroup-relative LDS offset
    unsigned ldsOff =
        (unsigned)(unsigned long long)(const void*)&sb[0] + (unsigned)lane * 16u;

    asm volatile("global_load_async_to_lds_b128 %0, %1, off"
                 :: "v"(ldsOff), "v"(gaddr) : "memory");
    asm volatile("s_wait_asynccnt 0" ::: "memory");

    // Per-lane area of own box (rows for lanes 0..15, cols for 16..31).
    float4 mine = sb[lane];
    float areaMine = (mine.z - mine.x) * (mine.w - mine.y);

    int n = lane & 15;
    float4 cb = sb[16 + n];                         // column box for this lane
    float areaColN = (cb.z - cb.x) * (cb.w - cb.y);

    // A: 16x4 f32  -> lanes 0..15 hold K=0,1 in v[0],v[1]; lanes 16..31 K=2,3 (zero)
    v2f aop = { (lane < 16) ? areaMine : 0.f, (lane < 16) ? 1.f : 0.f };
    // B: 4x16 f32  -> lanes 0..15 hold rows K=0 (=1) and K=1 (=area_col)
    v2f bop = { (lane < 16) ? 1.f : 0.f, (lane < 16) ? areaColN : 0.f };
    v8f csum = {};
    csum = __builtin_amdgcn_wmma_f32_16x16x4_f32(
        false, aop, false, bop, (short)0, csum, false, false);

    int hiSel = lane >> 4;
#pragma unroll
    for (int v = 0; v < 8; ++v) {
        int m = v + 8 * hiSel;
        float4 rb = sb[m];
        float ix1 = fmaxf(rb.x, cb.x), iy1 = fmaxf(rb.y, cb.y);
        float ix2 = fminf(rb.z, cb.z), iy2 = fminf(rb.w, cb.w);
        float inter = fmaxf(ix2 - ix1, 0.f) * fmaxf(iy2 - iy1, 0.f);
        float denom = fmaxf(csum[v] - inter, 1e-9f);   // area_m + area_n - inter
        bool pred = inter > NMS_TH * denom;            // == inter/denom > TH
        if (rt == ct) pred = pred && (n > m);          // only j > i suppresses
        unsigned long long bal = __ballot((int)pred);
        if (lane == 0) {
            supp[(unsigned)(rt * 16 + v) * (unsigned)TILES + (unsigned)ct] =
                (unsigned)(bal & 0xFFFFull);
            supp[(unsigned)(rt * 16 + v + 8) * (unsigned)TILES + (unsigned)ct] =
                (unsigned)((bal >> 16) & 0xFFFFull);
        }
    }
}

// ---------------------------------------------------------------------------
// Kernel 5: serial greedy reduction over the bitmask + compaction to output.
// removed[] = 375 words x 16 bits in LDS. Invalid (min-size) boxes start removed.
// ---------------------------------------------------------------------------
__global__ void nms_finalize_kernel(const unsigned* __restrict__ supp,
                                    const int* __restrict__ validFlags,
                                    const float4* __restrict__ topBoxes,
                                    float* __restrict__ out) {
    __shared__ unsigned rem[TILES];
    int t = threadIdx.x;

    for (int i = t; i < POST_TOPN * 5; i += blockDim.x) out[i] = 0.f;

    if (t < TILES) {
        unsigned bits = 0;
        for (int b = 0; b < 16; ++b)
            if (!validFlags[t * 16 + b]) bits |= (1u << b);
        rem[t] = bits;
    }
    __syncthreads();

    for (int i = 0; i < PRE_TOPN; ++i) {
        bool alive = ((rem[i >> 4] >> (i & 15)) & 1u) == 0u;
        __syncthreads();
        if (alive && t < TILES && t >= (i >> 4))
            rem[t] |= supp[(unsigned)i * (unsigned)TILES + (unsigned)t];
        __syncthreads();
    }

    if (t == 0) {
        int cnt = 0;
        for (int i = 0; i < PRE_TOPN && cnt < POST_TOPN; ++i) {
            if (!((rem[i >> 4] >> (i & 15)) & 1u)) {
                float4 b = topBoxes[i];
                out[cnt * 5 + 0] = 0.f;
                out[cnt * 5 + 1] = b.x;
                out[cnt * 5 + 2] = b.y;
                out[cnt * 5 + 3] = b.z;
                out[cnt * 5 + 4] = b.w;
                ++cnt;
            }
        }
    }
}

// ---------------------------------------------------------------------------
// workspace layout (bytes, all 16B aligned):
//   keys      : NPAD * 8              =  2,097,152
//   boxesAll  : N_PROP * 16           =  3,538,944
//   topBoxes  : 6000 * 16             =     96,000
//   validFlags: 6000 * 4              =     24,000
//   supp      : 6000 * 375 * 4        =  9,000,000   (total ~14.8 MB)
// ---------------------------------------------------------------------------
extern "C" void kernel_launch(void* const* d_in, const int* in_sizes, int n_in,
                              void* d_out, int out_size, void* d_ws, size_t ws_size,
                              hipStream_t stream) {
    (void)in_sizes; (void)n_in; (void)out_size; (void)ws_size;
    const float* scores = (const float*)d_in[0];
    const float* deltas = (const float*)d_in[1];
    const float* iminfo = (const float*)d_in[2];
    float* out = (float*)d_out;

    char* ws = (char*)d_ws;
    unsigned long long* keys = (unsigned long long*)(ws);
    float4* boxesAll = (float4*)(ws + 2097152);
    float4* topBoxes = (float4*)(ws + 5636096);
    int* validFlags  = (int*)(ws + 5732096);
    unsigned* supp   = (unsigned*)(ws + 5756096);

    prop_decode_kernel<<<NPAD / 256, 256, 0, stream>>>(scores, deltas, iminfo,
                                                       keys, boxesAll);

    // Bitonic sort: global steps only for compare distance >= 1024; all
    // lower stages of each phase fused into one LDS-resident kernel.
    for (unsigned k = 2; k <= (unsigned)NPAD; k <<= 1) {
        unsigned j = k >> 1;
        for (; j >= 1024u; j >>= 1)
            bitonic_step_kernel<<<NPAD / 256, 256, 0, stream>>>(keys, j, k);
        bitonic_fused_kernel<<<NPAD / 1024, 512, 0, stream>>>(keys, j, k);
    }

    gather_kernel<<<(PRE_TOPN + 255) / 256, 256, 0, stream>>>(keys, boxesAll,
                                                              topBoxes, validFlags);

    supp_tile_kernel<<<dim3(TILES, TILES), 32, 0, stream>>>(topBoxes, supp);

    nms_finalize_kernel<<<1, 384, 0, stream>>>(supp, validFlags, topBoxes, out);
}